// SoftArgmax2D_35871566856370
// MI455X (gfx1250) — compile-verified
//
#include <hip/hip_runtime.h>

typedef __attribute__((ext_vector_type(2))) float v2f;
typedef __attribute__((ext_vector_type(8))) float v8f;

#define TILE_H 128
#define TILE_W 128
#define TILE_HW (TILE_H * TILE_W)

// win(d) * exp(x - m) at window position (row r, col c) of the 16x16 tile
// centered so that (r,c) == (5,5) is the argmax pixel. Zero outside the
// Parzen support (d > 5) or outside the image.
__device__ __forceinline__ float parzen_exp(const float* __restrict__ p,
                                            float m, int ax, int ay,
                                            int r, int c) {
  float dx = (float)(c - 5);
  float dy = (float)(r - 5);
  float d2 = dx * dx + dy * dy;
  if (d2 > 25.0f) return 0.0f;                 // outside window support
  int gy = ay - 5 + r;
  int gx = ax - 5 + c;
  if (gy < 0 || gy >= TILE_H || gx < 0 || gx >= TILE_W) return 0.0f;
  float d  = __builtin_amdgcn_sqrtf(d2);        // v_sqrt_f32, exact-int input
  float dp = d * 0.2f;                          // d / hwidth, hwidth = 5
  float win;
  if (d <= 2.5f) {
    win = 1.0f - 6.0f * dp * dp * (1.0f - dp);  // near-mode cubic
  } else {
    float t = 1.0f - dp;
    win = 2.0f * t * t * t;                     // tail cubic
  }
  float v = p[gy * TILE_W + gx];
  return win * expf(v - m);                     // precise expf path
}

__global__ __launch_bounds__(256) void softargmax2d_kernel(
    const float* __restrict__ x, float* __restrict__ out, int nchan) {
  const int bc = blockIdx.x;
  if (bc >= nchan) return;
  const float* __restrict__ p = x + (size_t)bc * TILE_HW;
  const int tid  = threadIdx.x;
  const int lane = tid & 31;
  const int wave = tid >> 5;

  // ---------------- Phase 1: streaming max / argmax (memory-bound) --------
  float vmax = -__builtin_inff();
  int   imax = 0;
  const float4* p4 = (const float4*)p;
#pragma unroll
  for (int j = 0; j < 16; ++j) {
    int i4 = tid + 256 * j;          // coalesced b128 stream
    float4 v = p4[i4];
    int base = i4 * 4;
    if (v.x > vmax) { vmax = v.x; imax = base;     }
    if (v.y > vmax) { vmax = v.y; imax = base + 1; }
    if (v.z > vmax) { vmax = v.z; imax = base + 2; }
    if (v.w > vmax) { vmax = v.w; imax = base + 3; }
  }
  // wave32 reduction, first-occurrence tie-break (lowest index wins)
#pragma unroll
  for (int off = 16; off >= 1; off >>= 1) {
    float vo = __shfl_xor(vmax, off, 32);
    int   io = __shfl_xor(imax, off, 32);
    if (vo > vmax || (vo == vmax && io < imax)) { vmax = vo; imax = io; }
  }

  __shared__ float s_val[8];
  __shared__ int   s_idx[8];
  __shared__ float s_m;
  __shared__ int   s_am;
  if (lane == 0) { s_val[wave] = vmax; s_idx[wave] = imax; }
  __syncthreads();
  if (tid == 0) {
    float m = s_val[0]; int am = s_idx[0];
#pragma unroll
    for (int wv = 1; wv < 8; ++wv) {
      float vo = s_val[wv]; int io = s_idx[wv];
      if (vo > m || (vo == m && io < am)) { m = vo; am = io; }
    }
    s_m = m; s_am = am;
  }
  __syncthreads();

  // ---------------- Phase 2: windowed expectation via WMMA marginals ------
  // S[r][c] (16x16, zero-padded 11x11 window). Softmax denominator cancels
  // under renormalization, so only win*exp(x-max) sums are needed.
  if (wave == 0) {
    const float m  = s_m;
    const int   am = s_am;
    const int   ax = am & (TILE_W - 1);
    const int   ay = am >> 7;
    const int   pidx  = lane & 15;            // M (for A) / N (for B)
    const int   qbase = (lane < 16) ? 0 : 2;  // K sub-slot per lane half

    v8f accA = {};  // D_A[m][n] = R[m]  (row marginals, bcast over n)
    v8f accB = {};  // D_B[m][n] = Cm[n] (col marginals, bcast over m)
    v2f ones; ones[0] = 1.0f; ones[1] = 1.0f;

#pragma unroll
    for (int j = 0; j < 4; ++j) {            // K chunks k0 = 4j
      int q0 = 4 * j + qbase;
      v2f a;                                  // A(16x4): S[pidx][q0], S[pidx][q0+1]
      a[0] = parzen_exp(p, m, ax, ay, pidx, q0);
      a[1] = parzen_exp(p, m, ax, ay, pidx, q0 + 1);
      v2f b;                                  // B(4x16): S[q0][pidx], S[q0+1][pidx]
      b[0] = parzen_exp(p, m, ax, ay, q0, pidx);
      b[1] = parzen_exp(p, m, ax, ay, q0 + 1, pidx);
      accA = __builtin_amdgcn_wmma_f32_16x16x4_f32(
          false, a, false, ones, (short)0, accA, false, false);
      accB = __builtin_amdgcn_wmma_f32_16x16x4_f32(
          false, ones, false, b, (short)0, accB, false, false);
    }

    // C/D layout: VGPR i, lanes 0-15 -> M=i, lanes 16-31 -> M=8+i.
    float Wp = 0.0f, Yp = 0.0f;
    const int mbase = (lane < 16) ? 0 : 8;
#pragma unroll
    for (int i = 0; i < 8; ++i) {
      float r = accA[i];                      // R[mbase+i]
      Wp += r;
      Yp += r * (float)(ay - 5 + mbase + i);  // y coordinate weight
    }
    Wp += __shfl_xor(Wp, 16, 32);
    Yp += __shfl_xor(Yp, 16, 32);

    // accB[0]: lane l holds Cm[l % 16] (duplicated across halves)
    float Xp = accB[0] * (float)(ax - 5 + pidx);
    Xp += __shfl_xor(Xp, 1, 32);
    Xp += __shfl_xor(Xp, 2, 32);
    Xp += __shfl_xor(Xp, 4, 32);
    Xp += __shfl_xor(Xp, 8, 32);

    if (lane == 0) {
      out[2 * bc + 0] = Xp / Wp;              // x coordinate
      out[2 * bc + 1] = Yp / Wp;              // y coordinate
    }
  }
}

extern "C" void kernel_launch(void* const* d_in, const int* in_sizes, int n_in,
                              void* d_out, int out_size, void* d_ws, size_t ws_size,
                              hipStream_t stream) {
  (void)n_in; (void)out_size; (void)d_ws; (void)ws_size;
  const float* x = (const float*)d_in[0];
  float* out = (float*)d_out;
  int nchan = in_sizes[0] / TILE_HW;          // B*C = 2048
  softargmax2d_kernel<<<nchan, 256, 0, stream>>>(x, out, nchan);
}